// GNN_76338748719223
// MI455X (gfx1250) — compile-verified
//
#include <hip/hip_runtime.h>

#define HID       128
#define LDS_PITCH 132   // 128 + 4 floats padding: avoids 16-way LDS bank conflicts
#define MT        80    // M rows per block (100000 = 1250 * 80, exact)
#define MSUB      5     // 16-row WMMA subtiles per block

typedef __attribute__((ext_vector_type(2))) float v2f;
typedef __attribute__((ext_vector_type(8))) float v8f;

// ---------------------------------------------------------------------------
// Zero a float buffer
// ---------------------------------------------------------------------------
__global__ void sage_zero(float* __restrict__ p, int n) {
    int i = blockIdx.x * blockDim.x + threadIdx.x;
    if (i < n) p[i] = 0.0f;
}

// ---------------------------------------------------------------------------
// rdeg[i] = 1 / max(deg[i], 1)
// ---------------------------------------------------------------------------
__global__ void sage_rdeg(const float* __restrict__ degf,
                          float* __restrict__ rdeg, int n) {
    int i = blockIdx.x * blockDim.x + threadIdx.x;
    if (i < n) {
        float d = degf[i];
        rdeg[i] = d > 1.0f ? 1.0f / d : 1.0f;
    }
}

// ---------------------------------------------------------------------------
// Edge scatter: one wave32 per edge. Lane l accumulates features l, l+32,
// l+64, l+96 of x[src] into agg[dst] with global_atomic_add_f32.
// Lane 0 bumps the degree counter (layer 1 only; degrees identical per layer).
// ---------------------------------------------------------------------------
__global__ void sage_scatter(const float* __restrict__ x,
                             const int*   __restrict__ src,
                             const int*   __restrict__ dst,
                             float*       __restrict__ agg,
                             float*       __restrict__ degf,
                             int nEdges, int doDeg) {
    int gtid = blockIdx.x * blockDim.x + threadIdx.x;
    int edge = gtid >> 5;
    int lane = gtid & 31;
    if (edge >= nEdges) return;

    int s = src[edge];
    int d = dst[edge];
    const float* xs = x   + (size_t)s * HID;
    float*       ad = agg + (size_t)d * HID;

    atomicAdd(ad + lane,      xs[lane]);
    atomicAdd(ad + lane + 32, xs[lane + 32]);
    atomicAdd(ad + lane + 64, xs[lane + 64]);
    atomicAdd(ad + lane + 96, xs[lane + 96]);
    if (doDeg && lane == 0) atomicAdd(degf + d, 1.0f);
}

// ---------------------------------------------------------------------------
// Fused SAGE dense stage:  out = relu?( mean @ Wl + bl + x @ Wr )
// Block = 256 threads = 8 waves, owns an 80-row node tile (5 x 16 subtiles).
// Wave w owns output columns [16w, 16w+16). Each B register pair (one 4x16
// weight sliver) feeds 5 WMMAs -> 5 independent accumulator chains keep the
// fp32 matrix pipe busy and cut weight traffic 5x.
// ---------------------------------------------------------------------------
__global__ void sage_gemm(const float* __restrict__ agg,
                          const float* __restrict__ rdeg,
                          const float* __restrict__ xin,
                          const float* __restrict__ wl,
                          const float* __restrict__ bl,
                          const float* __restrict__ wr,
                          float*       __restrict__ out,
                          int applyRelu) {
    __shared__ float sMean[MT * LDS_PITCH];
    __shared__ float sX   [MT * LDS_PITCH];

    const int tid      = threadIdx.x;         // 0..255
    const int tileBase = blockIdx.x * MT;     // first node row of this tile

    // Cooperative stage: MT rows x 128 floats per matrix = MT*32 float4 each.
    for (int i = tid; i < MT * 32; i += 256) {
        int r   = i >> 5;                 // row in tile
        int c4  = (i & 31) << 2;          // starting column
        int row = tileBase + r;
        float rd = rdeg[row];
        float4 av = *(const float4*)(agg + (size_t)row * HID + c4);
        float4 mv = make_float4(av.x * rd, av.y * rd, av.z * rd, av.w * rd);
        *(float4*)(sMean + r * LDS_PITCH + c4) = mv;
        *(float4*)(sX    + r * LDS_PITCH + c4) =
            *(const float4*)(xin + (size_t)row * HID + c4);
    }
    __syncthreads();

    const int wave  = tid >> 5;              // 0..7 -> N tile
    const int lane  = tid & 31;
    const int lrow  = lane & 15;             // A row / B column within subtile
    const int halfK = (lane >> 4) << 1;      // 0 for lanes 0-15, 2 for 16-31
    const int nCol  = wave * 16 + lrow;      // global output column

    v8f acc[MSUB];
    #pragma unroll
    for (int mt = 0; mt < MSUB; ++mt) acc[mt] = (v8f){};

    // mean @ Wl : each 4x16 weight sliver reused across 5 M subtiles
    for (int k0 = 0; k0 < HID; k0 += 4) {
        v2f b;
        b.x = wl[(size_t)(k0 + halfK)     * HID + nCol];
        b.y = wl[(size_t)(k0 + halfK + 1) * HID + nCol];
        #pragma unroll
        for (int mt = 0; mt < MSUB; ++mt) {
            const float* ap = sMean + (mt * 16 + lrow) * LDS_PITCH + k0 + halfK;
            v2f a; a.x = ap[0]; a.y = ap[1];
            acc[mt] = __builtin_amdgcn_wmma_f32_16x16x4_f32(
                          false, a, false, b, (short)0, acc[mt], false, false);
        }
    }

    // x @ Wr : same accumulators
    for (int k0 = 0; k0 < HID; k0 += 4) {
        v2f b;
        b.x = wr[(size_t)(k0 + halfK)     * HID + nCol];
        b.y = wr[(size_t)(k0 + halfK + 1) * HID + nCol];
        #pragma unroll
        for (int mt = 0; mt < MSUB; ++mt) {
            const float* ap = sX + (mt * 16 + lrow) * LDS_PITCH + k0 + halfK;
            v2f a; a.x = ap[0]; a.y = ap[1];
            acc[mt] = __builtin_amdgcn_wmma_f32_16x16x4_f32(
                          false, a, false, b, (short)0, acc[mt], false, false);
        }
    }

    // Epilogue: bias + optional ReLU + store.
    // D layout: vgpr r, lane L -> row = r + (L<16 ? 0 : 8), col = L&15.
    const float bias   = bl[nCol];
    const int   rowOff = (lane < 16) ? 0 : 8;
    #pragma unroll
    for (int mt = 0; mt < MSUB; ++mt) {
        #pragma unroll
        for (int r = 0; r < 8; ++r) {
            float v = acc[mt][r] + bias;
            if (applyRelu) v = v > 0.0f ? v : 0.0f;
            out[(size_t)(tileBase + mt * 16 + r + rowOff) * HID + nCol] = v;
        }
    }
}

// ---------------------------------------------------------------------------
// Driver
// ---------------------------------------------------------------------------
extern "C" void kernel_launch(void* const* d_in, const int* in_sizes, int n_in,
                              void* d_out, int out_size, void* d_ws, size_t ws_size,
                              hipStream_t stream) {
    const float* x    = (const float*)d_in[0];
    const int*   ei   = (const int*)  d_in[1];   // [2, E] flat: src then dst
    const float* w1l  = (const float*)d_in[2];
    const float* b1l  = (const float*)d_in[3];
    const float* w1r  = (const float*)d_in[4];
    const float* w2l  = (const float*)d_in[5];
    const float* b2l  = (const float*)d_in[6];
    const float* w2r  = (const float*)d_in[7];

    const int N = in_sizes[0] / HID;     // 100000
    const int E = in_sizes[1] / 2;       // 600000
    const int* src = ei;
    const int* dst = ei + E;

    // workspace carve-up: agg | deg (contiguous for the layer-1 zero pass),
    // then h (layer-1 activations) and rdeg.
    float* agg  = (float*)d_ws;                     // N*128 floats
    float* deg  = agg  + (size_t)N * HID;           // N floats
    float* h    = deg  + N;                         // N*128 floats
    float* rdeg = h    + (size_t)N * HID;           // N floats
    float* out  = (float*)d_out;

    const int zeroAllN   = N * (HID + 1);
    const int zeroAggN   = N * HID;
    const int zAllGrid   = (zeroAllN + 255) / 256;
    const int zAggGrid   = (zeroAggN + 255) / 256;
    const int nGrid      = (N + 255) / 256;
    const int scatGrid   = (int)(((long long)E * 32 + 255) / 256);
    const int gemmGrid   = N / MT;                  // 1250 (N % 80 == 0)

    // ---- layer 1 ----
    sage_zero   <<<zAllGrid, 256, 0, stream>>>(agg, zeroAllN);        // agg + deg
    sage_scatter<<<scatGrid, 256, 0, stream>>>(x, src, dst, agg, deg, E, 1);
    sage_rdeg   <<<nGrid,    256, 0, stream>>>(deg, rdeg, N);
    sage_gemm   <<<gemmGrid, 256, 0, stream>>>(agg, rdeg, x, w1l, b1l, w1r, h, 1);

    // ---- layer 2 (degrees unchanged: reuse rdeg, zero agg only) ----
    sage_zero   <<<zAggGrid, 256, 0, stream>>>(agg, zeroAggN);
    sage_scatter<<<scatGrid, 256, 0, stream>>>(h, src, dst, agg, deg, E, 0);
    sage_gemm   <<<gemmGrid, 256, 0, stream>>>(agg, rdeg, h, w2l, b2l, w2r, out, 0);
}